// HingeTreeConv1d_69114613729301
// MI455X (gfx1250) — compile-verified
//
#include <hip/hip_runtime.h>
#include <hip/hip_bf16.h>
#include <float.h>

// Problem constants (from reference)
#define B_      8
#define C_IN_   16
#define L_      4096
#define C_OUT_  32
#define DEPTH_  5
#define K_      7
#define PAD_    3
#define NINNER_ 31
#define NLEAF_  32

// Tiling
#define TILE    256                 // l positions per block
#define XVALID  (TILE + 2*PAD_)     // 262 staged x elems per channel row
#define XS      264                 // padded LDS row stride (floats)
#define NTHR    256

typedef float v8f __attribute__((ext_vector_type(8)));
typedef float v2f __attribute__((ext_vector_type(2)));

#ifndef __has_builtin
#define __has_builtin(x) 0
#endif

#if __has_builtin(__builtin_amdgcn_global_load_async_to_lds_b32)
#define USE_ASYNC_LDS 1
#else
#define USE_ASYNC_LDS 0
#endif

#if __has_builtin(__builtin_amdgcn_wmma_f32_16x16x4_f32)
#define USE_WMMA 1
#else
#define USE_WMMA 0
#endif

__global__ __launch_bounds__(NTHR)
void hinge_tree_conv1d_kernel(const float* __restrict__ x,
                              const float* __restrict__ thr,
                              const int*   __restrict__ ordn,
                              const float* __restrict__ wts,
                              float*       __restrict__ out)
{
    __shared__ float sx  [C_IN_ * XS];   // padded x window tile
    __shared__ int2  s_tn[C_IN_ * 32];   // packed {threshold bits, ordinal} per node
    __shared__ float s_w [C_IN_ * 32];   // leaf weights

    const int tid = threadIdx.x;
    const int l0  = blockIdx.x * TILE;
    const int o   = blockIdx.y;
    const int b   = blockIdx.z;

    // ---- stage tree parameters for this output channel (all 16 input channels) ----
    for (int idx = tid; idx < C_IN_ * 32; idx += NTHR) {
        const int c = idx >> 5, n = idx & 31;
        s_w[idx] = wts[(o * C_IN_ + c) * NLEAF_ + n];
        float tv = 0.f; int ov = 0;
        if (n < NINNER_) {
            tv = thr [(o * C_IN_ + c) * NINNER_ + n];
            ov = ordn[(o * C_IN_ + c) * NINNER_ + n];
        }
        s_tn[idx] = make_int2(__float_as_int(tv), ov);   // one ds_load_b64 per tree level
    }

    // ---- stage x tile (with zero padding at sequence edges) ----
    for (int idx = tid; idx < C_IN_ * XS; idx += NTHR) {
        const int c = idx / XS;
        const int i = idx - c * XS;
        const int g = l0 - PAD_ + i;
        const bool valid = (i < XVALID) && (g >= 0) && (g < L_);
#if USE_ASYNC_LDS
        if (valid) {
            float* gsrc = const_cast<float*>(x) + ((size_t)(b * C_IN_ + c)) * L_ + g;
            __builtin_amdgcn_global_load_async_to_lds_b32(
                (__attribute__((address_space(1))) int*)(void*)gsrc,
                (__attribute__((address_space(3))) int*)(void*)&sx[idx],
                0, 0);
        } else {
            sx[idx] = 0.f;
        }
#else
        sx[idx] = valid ? x[((size_t)(b * C_IN_ + c)) * L_ + g] : 0.f;
#endif
    }
#if USE_ASYNC_LDS
    asm volatile("s_wait_asynccnt 0" ::: "memory");
#endif
    __syncthreads();

    const int lane = tid & 31;
    const int wv   = tid >> 5;
    const int m    = lane & 15;   // row (l offset within 16-l subtile)
    const int half = lane >> 4;   // which half of the c-range this lane handles

    float* __restrict__ outrow = out + ((size_t)(b * C_OUT_ + o)) * L_;

    // Each of 8 waves handles two 16-l subtiles -> 256 l per block.
    for (int sub = 0; sub < 2; ++sub) {
        const int lbase = (wv * 2 + sub) * 16;  // subtile base within tile
        const int ll    = lbase + m;            // this lane's l within tile

        // Tree traversal: lane handles 8 of 16 channels, arranged to match the
        // 16x4 f32 WMMA A-matrix layout (lanes 0-15: K=0,1; lanes 16-31: K=2,3).
        float mw[8];
#pragma unroll
        for (int cc = 0; cc < 8; ++cc) {
            const int c = (cc >> 1) * 4 + half * 2 + (cc & 1);
            const float* __restrict__ xr = &sx[c * XS];
            const int2*  __restrict__ tn = &s_tn[c * 32];
            int   node = 0;
            float marg = FLT_MAX;
#pragma unroll
            for (int d = 0; d < DEPTH_; ++d) {
                const int2  p = tn[node];                 // ds_load_b64: {thr, ord}
                const float t = __int_as_float(p.x);
                const float f = xr[ll + p.y];             // ds_load_b32 gather
                const float diff = f - t;
                marg = fminf(marg, fabsf(diff));
                node = 2 * node + 1 + (diff > 0.f ? 1 : 0);
            }
            mw[cc] = marg * s_w[c * 32 + (node - NINNER_)];
        }

#if USE_WMMA
        // Reduce over c with an exact f32 WMMA matvec against all-ones:
        // D[m,n] = sum_k A[m,k] * 1  (every column n holds out[lbase+m]).
        v8f acc = {0.f, 0.f, 0.f, 0.f, 0.f, 0.f, 0.f, 0.f};
        v2f bones; bones[0] = 1.f; bones[1] = 1.f;
#pragma unroll
        for (int ch = 0; ch < 4; ++ch) {
            v2f a; a[0] = mw[2 * ch]; a[1] = mw[2 * ch + 1];
            acc = __builtin_amdgcn_wmma_f32_16x16x4_f32(
                false, a, false, bones, (short)0, acc, false, false);
        }
        // Column N=0 lives in lanes 0 (rows 0..7) and 16 (rows 8..15).
        if (m == 0) {
            float* p = outrow + l0 + lbase + half * 8;
            *(float4*)(p)     = make_float4(acc[0], acc[1], acc[2], acc[3]);
            *(float4*)(p + 4) = make_float4(acc[4], acc[5], acc[6], acc[7]);
        }
#else
        float sum = 0.f;
#pragma unroll
        for (int cc = 0; cc < 8; ++cc) sum += mw[cc];
        sum += __shfl_xor(sum, 16, 32);   // combine the two c-halves
        if (half == 0) outrow[l0 + ll] = sum;
#endif
    }
}

extern "C" void kernel_launch(void* const* d_in, const int* in_sizes, int n_in,
                              void* d_out, int out_size, void* d_ws, size_t ws_size,
                              hipStream_t stream) {
    (void)in_sizes; (void)n_in; (void)out_size; (void)d_ws; (void)ws_size;
    const float* x    = (const float*)d_in[0];
    const float* thrp = (const float*)d_in[1];
    const int*   ordp = (const int*)d_in[2];
    const float* wtsp = (const float*)d_in[3];
    float*       outp = (float*)d_out;

    dim3 grid(L_ / TILE, C_OUT_, B_);
    dim3 block(NTHR);
    hipLaunchKernelGGL(hinge_tree_conv1d_kernel, grid, block, 0, stream,
                       x, thrp, ordp, wtsp, outp);
}